// deFmNu_53068615910202
// MI455X (gfx1250) — compile-verified
//
#include <hip/hip_runtime.h>
#include <math.h>

typedef __attribute__((ext_vector_type(2))) float v2f;
typedef __attribute__((ext_vector_type(8))) float v8f;

#define NF    26      // sparse fields
#define ED    4       // embedding dim
#define NUMD  13      // dense features
#define KDIM  117     // F*E + NUM
#define KPAD  120     // padded K (multiple of 4)
#define DSTR  124     // deep LDS row stride (words), 16B aligned, lower bank-conflict
#define L1N   12
#define L2N   8
#define L3N   4
#define ROWS  256     // rows per block == threads per block
#define EPSV  1e-5f

__global__ __launch_bounds__(256) void deepfm_fused_kernel(
    const int*   __restrict__ x1,   const float* __restrict__ x2,
    const float* __restrict__ fw1,  const float* __restrict__ fw2,
    const float* __restrict__ bias,
    const float* __restrict__ w1, const float* __restrict__ b1,
    const float* __restrict__ g1, const float* __restrict__ be1,
    const float* __restrict__ m1, const float* __restrict__ v1,
    const float* __restrict__ w2, const float* __restrict__ b2,
    const float* __restrict__ g2, const float* __restrict__ be2,
    const float* __restrict__ m2, const float* __restrict__ v2,
    const float* __restrict__ w3, const float* __restrict__ b3,
    const float* __restrict__ g3, const float* __restrict__ be3,
    const float* __restrict__ m3, const float* __restrict__ v3,
    float* __restrict__ out)
{
    __shared__ float s_deep[ROWS][DSTR];   // [row][k] activations, fp32
    __shared__ float s_w1t[16][DSTR];      // w1 transposed: [n][k], zero padded
    __shared__ float s_h1[ROWS][L1N];
    __shared__ float s_w2[L1N][L2N];
    __shared__ float s_w3[L2N][L3N];
    __shared__ float sA1[16], sC1[16];     // folded bias+BN: y = x*A + C
    __shared__ float sA2[L2N], sC2[L2N];
    __shared__ float sA3[L3N], sC3[L3N];

    const int tid = threadIdx.x;
    const int row = blockIdx.x * ROWS + tid;

    // ---------------- phase 0: preload weights / fold BN constants ----------
    for (int i = tid; i < 16 * DSTR; i += 256) {
        int n = i / DSTR, k = i % DSTR;
        s_w1t[n][k] = (n < L1N && k < KDIM) ? w1[k * L1N + n] : 0.0f;
    }
    if (tid < L1N * L2N) s_w2[tid / L2N][tid % L2N] = w2[tid];
    if (tid < L2N * L3N) s_w3[tid / L3N][tid % L3N] = w3[tid];
    if (tid < 16) {
        if (tid < L1N) {
            float sc = g1[tid] * rsqrtf(v1[tid] + EPSV);
            sA1[tid] = sc;
            sC1[tid] = (b1[tid] - m1[tid]) * sc + be1[tid];
        } else { sA1[tid] = 0.0f; sC1[tid] = 0.0f; }
    }
    if (tid >= 32 && tid < 32 + L2N) {
        int j = tid - 32;
        float sc = g2[j] * rsqrtf(v2[j] + EPSV);
        sA2[j] = sc; sC2[j] = (b2[j] - m2[j]) * sc + be2[j];
    }
    if (tid >= 64 && tid < 64 + L3N) {
        int j = tid - 64;
        float sc = g3[j] * rsqrtf(v3[j] + EPSV);
        sA3[j] = sc; sC3[j] = (b3[j] - m3[j]) * sc + be3[j];
    }

    // ---------------- phase 1: gathers + FM terms + stage deep matrix -------
    float fm1 = 0.0f;
    float s0 = 0.f, s1 = 0.f, s2 = 0.f, s3 = 0.f;
    float q0 = 0.f, q1 = 0.f, q2 = 0.f, q3 = 0.f;
    {
        const float4* emb = (const float4*)fw2;   // E=4 -> 16B aligned rows
        #pragma unroll
        for (int f = 0; f < NF; ++f) {
            int idx = x1[row * NF + f];
            fm1 += fw1[idx];
            float4 e = emb[idx];
            s0 += e.x; s1 += e.y; s2 += e.z; s3 += e.w;
            q0 += e.x * e.x; q1 += e.y * e.y; q2 += e.z * e.z; q3 += e.w * e.w;
            *(float4*)&s_deep[tid][f * 4] = e;
        }
        #pragma unroll
        for (int j = 0; j < NUMD; ++j) s_deep[tid][NF * 4 + j] = x2[row * NUMD + j];
        s_deep[tid][117] = 0.0f; s_deep[tid][118] = 0.0f; s_deep[tid][119] = 0.0f;
    }
    const float fm2 = 0.5f * ((s0 * s0 - q0) + (s1 * s1 - q1) +
                              (s2 * s2 - q2) + (s3 * s3 - q3));

    __syncthreads();

    // ---------------- phase 2: layer-1 GEMM via V_WMMA_F32_16X16X4_F32 ------
    // A (16x4 f32): lanes 0-15 hold row M=lane, {K0,K1}; lanes 16-31 {K2,K3}.
    // B (4x16 f32): lanes 0-15 hold col N=lane, {K0,K1}; lanes 16-31 {K2,K3}.
    {
        const int lane = tid & 31;
        const int wave = tid >> 5;
        const int half = lane >> 4;   // selects K pair within the chunk of 4
        const int mn   = lane & 15;   // A row / B col
        #pragma unroll
        for (int t = 0; t < 2; ++t) {
            const int tileBase = (wave * 2 + t) * 16;
            v8f acc = {};
            #pragma unroll
            for (int c = 0; c < KPAD / 4; ++c) {
                const int k = c * 4 + half * 2;
                v2f a = *(const v2f*)&s_deep[tileBase + mn][k];
                v2f b = *(const v2f*)&s_w1t[mn][k];
                acc = __builtin_amdgcn_wmma_f32_16x16x4_f32(
                    false, a, false, b, (short)0, acc, false, false);
            }
            // folded bias+BN + ReLU, write h1 (cols >= 12 are padding)
            if (mn < L1N) {
                const float sc = sA1[mn], sh = sC1[mn];
                #pragma unroll
                for (int i = 0; i < 8; ++i) {
                    const int r = tileBase + i + half * 8;
                    float h = acc[i] * sc + sh;
                    s_h1[r][mn] = h > 0.0f ? h : 0.0f;
                }
            }
        }
    }

    __syncthreads();

    // ---------------- phase 3: layers 2 & 3 + reduce + sigmoid --------------
    float h2[L2N];
    #pragma unroll
    for (int j = 0; j < L2N; ++j) {
        float a = 0.0f;
        #pragma unroll
        for (int k = 0; k < L1N; ++k) a += s_h1[tid][k] * s_w2[k][j];
        a = a * sA2[j] + sC2[j];
        h2[j] = a > 0.0f ? a : 0.0f;
    }
    float hsum = 0.0f;
    #pragma unroll
    for (int j = 0; j < L3N; ++j) {
        float a = 0.0f;
        #pragma unroll
        for (int k = 0; k < L2N; ++k) a += h2[k] * s_w3[k][j];
        a = a * sA3[j] + sC3[j];
        hsum += a > 0.0f ? a : 0.0f;
    }
    const float z = fm1 + fm2 + hsum + bias[0];
    out[row] = 1.0f / (1.0f + __expf(-z));
}

extern "C" void kernel_launch(void* const* d_in, const int* in_sizes, int n_in,
                              void* d_out, int out_size, void* d_ws, size_t ws_size,
                              hipStream_t stream) {
    const int*   x1   = (const int*)d_in[0];
    const float* x2   = (const float*)d_in[1];
    const float* fw1  = (const float*)d_in[2];
    const float* fw2  = (const float*)d_in[3];
    const float* bias = (const float*)d_in[4];
    const float* w1 = (const float*)d_in[5];  const float* b1 = (const float*)d_in[6];
    const float* g1 = (const float*)d_in[7];  const float* be1 = (const float*)d_in[8];
    const float* m1 = (const float*)d_in[9];  const float* v1 = (const float*)d_in[10];
    const float* w2 = (const float*)d_in[11]; const float* b2 = (const float*)d_in[12];
    const float* g2 = (const float*)d_in[13]; const float* be2 = (const float*)d_in[14];
    const float* m2 = (const float*)d_in[15]; const float* v2 = (const float*)d_in[16];
    const float* w3 = (const float*)d_in[17]; const float* b3 = (const float*)d_in[18];
    const float* g3 = (const float*)d_in[19]; const float* be3 = (const float*)d_in[20];
    const float* m3 = (const float*)d_in[21]; const float* v3 = (const float*)d_in[22];

    const int rows = in_sizes[0] / NF;        // 16384
    const int grid = rows / ROWS;             // 64 blocks of 256 threads

    deepfm_fused_kernel<<<grid, ROWS, 0, stream>>>(
        x1, x2, fw1, fw2, bias,
        w1, b1, g1, be1, m1, v1,
        w2, b2, g2, be2, m2, v2,
        w3, b3, g3, be3, m3, v3,
        (float*)d_out);
}